// YOLOv1Loss_37469294691111
// MI455X (gfx1250) — compile-verified
//
#include <hip/hip_runtime.h>
#include <hip/hip_bf16.h>

typedef __attribute__((ext_vector_type(2))) float v2f;
typedef __attribute__((ext_vector_type(8))) float v8f;

#define YOLO_N 90          // 5*B + C
#define NCELLS (4096*7*7)  // 200704
#define NBLOCKS 784        // NCELLS / 256

// ---------------------------------------------------------------------------
// Pass 1: one cell per thread, streaming non-temporal loads, per-block reduce.
// ---------------------------------------------------------------------------
__global__ __launch_bounds__(256) void yolo_cell_kernel(
    const float* __restrict__ pred, const float* __restrict__ targ,
    float* __restrict__ block_sums, int ncells)
{
    __shared__ float red[256];
    const float inv_s = 1.0f / 7.0f;
    const int tid = threadIdx.x;
    float acc = 0.0f;

    for (int cell = blockIdx.x * 256 + tid; cell < ncells; cell += gridDim.x * 256) {
        const float* p = pred + (size_t)cell * YOLO_N;
        const float* t = targ + (size_t)cell * YOLO_N;

        // first 10 floats of each tensor (boxes + confs), 8-byte aligned
        float pb[10], tb[10];
        const v2f* p2 = (const v2f*)p;
        const v2f* t2 = (const v2f*)t;
        #pragma unroll
        for (int k = 0; k < 5; ++k) {
            v2f a = __builtin_nontemporal_load(p2 + k);
            v2f b = __builtin_nontemporal_load(t2 + k);
            pb[2*k] = a.x; pb[2*k+1] = a.y;
            tb[2*k] = b.x; tb[2*k+1] = b.y;
        }

        const float conf_t  = tb[4];
        const float coord_m = (conf_t > 0.0f)  ? 1.0f : 0.0f;
        const float noobj_m = (conf_t == 0.0f) ? 1.0f : 0.0f;

        const float d4 = pb[4] - tb[4], d9 = pb[9] - tb[9];
        const float dconf = d4*d4 + d9*d9;

        // pred box transforms (pa = xy*inv_s - wh/2 ; pb2 = pa*inv_s + wh/2)
        const float pax0 = pb[0]*inv_s - 0.5f*pb[2];
        const float pay0 = pb[1]*inv_s - 0.5f*pb[3];
        const float pbx0 = pax0*inv_s + 0.5f*pb[2];
        const float pby0 = pay0*inv_s + 0.5f*pb[3];
        const float pax1 = pb[5]*inv_s - 0.5f*pb[7];
        const float pay1 = pb[6]*inv_s - 0.5f*pb[8];
        const float pbx1 = pax1*inv_s + 0.5f*pb[7];
        const float pby1 = pay1*inv_s + 0.5f*pb[8];

        // target box 0 transform
        const float tax = tb[0]*inv_s - 0.5f*tb[2];
        const float tay = tb[1]*inv_s - 0.5f*tb[3];
        const float tbx = tax*inv_s + 0.5f*tb[2];
        const float tby = tay*inv_s + 0.5f*tb[3];
        const float t_area = (tbx - tax) * (tby - tay);

        // IoU for both pred boxes vs target box 0
        float tlx = fmaxf(pax0, tax), tly = fmaxf(pay0, tay);
        float brx = fminf(pbx0, tbx), bry = fminf(pby0, tby);
        float iw  = fmaxf(brx - tlx, 0.0f), ih = fmaxf(bry - tly, 0.0f);
        float inter = iw * ih;
        float parea0 = (pbx0 - pax0) * (pby0 - pay0);
        const float iou0 = inter / (parea0 + t_area - inter);

        tlx = fmaxf(pax1, tax); tly = fmaxf(pay1, tay);
        brx = fminf(pbx1, tbx); bry = fminf(pby1, tby);
        iw  = fmaxf(brx - tlx, 0.0f); ih = fmaxf(bry - tly, 0.0f);
        inter = iw * ih;
        float parea1 = (pbx1 - pax1) * (pby1 - pay1);
        const float iou1 = inter / (parea1 + t_area - inter);

        // argmax (first max wins => strict >)
        const bool sel1   = (iou1 > iou0);
        const float maxiou = sel1 ? iou1 : iou0;

        const float psx = sel1 ? pax1  : pax0;
        const float psy = sel1 ? pay1  : pay0;
        const float psw = sel1 ? pbx1  : pbx0;
        const float psh = sel1 ? pby1  : pby0;
        const float psc = sel1 ? pb[9] : pb[4];

        // tsel: idx==0 -> transformed target; idx==1 -> raw target box 1
        const float tsx = sel1 ? tb[5] : tax;
        const float tsy = sel1 ? tb[6] : tay;
        const float tsw = sel1 ? tb[7] : tbx;
        const float tsh = sel1 ? tb[8] : tby;

        const float dx = psx - tsx, dy = psy - tsy;
        const float dw = psw - tsw, dh = psh - tsh;
        const float dxywh = dx*dx + dy*dy + dw*dw + dh*dh;
        const float dc = psc - maxiou;

        // class loss over last 80 channels (offset 40B -> still 8-byte aligned)
        const v2f* pc = (const v2f*)(p + 10);
        const v2f* tc = (const v2f*)(t + 10);
        float cls0 = 0.0f, cls1 = 0.0f;
        #pragma unroll 8
        for (int k = 0; k < 40; ++k) {
            v2f a = __builtin_nontemporal_load(pc + k);
            v2f b = __builtin_nontemporal_load(tc + k);
            const float e0 = a.x - b.x, e1 = a.y - b.y;
            cls0 += e0 * e0;
            cls1 += e1 * e1;
        }

        acc += coord_m * (5.0f * dxywh + dc * dc + (cls0 + cls1))
             + 0.5f * noobj_m * dconf;
    }

    // deterministic block tree-reduction through LDS
    red[tid] = acc;
    __syncthreads();
    #pragma unroll
    for (int s = 128; s > 0; s >>= 1) {
        if (tid < s) red[tid] += red[tid + s];
        __syncthreads();
    }
    if (tid == 0) block_sums[blockIdx.x] = red[0];
}

// ---------------------------------------------------------------------------
// Pass 2: deterministic final reduction. 256 strided partials -> LDS ->
// 4 accumulating V_WMMA_F32_16X16X4_F32 with an all-ones B matrix:
// D = A*ones + C makes each D column the f32-exact row-sums of A.
// Column 0 of D lives in lane 0 (rows 0..7) and lane 16 (rows 8..15).
// ---------------------------------------------------------------------------
__global__ __launch_bounds__(256) void yolo_reduce_kernel(
    const float* __restrict__ block_sums, int nb, float* __restrict__ out)
{
    __shared__ float vals[256];
    const int tid = threadIdx.x;
    float s = 0.0f;
    for (int i = tid; i < nb; i += 256) s += block_sums[i];
    vals[tid] = s;
    __syncthreads();

    if (tid < 32) {  // wave 0, EXEC all ones — required for WMMA
        const v2f ones = {1.0f, 1.0f};  // 4x16 B matrix of all ones
        v8f c = {};
        #pragma unroll
        for (int q = 0; q < 4; ++q) {
            v2f a;
            a.x = vals[q * 64 + 2 * tid];
            a.y = vals[q * 64 + 2 * tid + 1];
            // (neg_a, A, neg_b, B, c_mod, C, reuse_a, reuse_b)
            c = __builtin_amdgcn_wmma_f32_16x16x4_f32(
                    false, a, false, ones, (short)0, c, false, false);
        }
        // per-lane sum over the 8 D VGPRs = sum of this lane's rows (col = lane%16)
        float r = c[0] + c[1] + c[2] + c[3] + c[4] + c[5] + c[6] + c[7];
        const float r_lo = __shfl(r, 0, 32);   // rows 0..7,  col 0
        const float r_hi = __shfl(r, 16, 32);  // rows 8..15, col 0
        if (tid == 0) out[0] = (r_lo + r_hi) * (1.0f / 4096.0f);
    }
}

extern "C" void kernel_launch(void* const* d_in, const int* in_sizes, int n_in,
                              void* d_out, int out_size, void* d_ws, size_t ws_size,
                              hipStream_t stream) {
    const float* pred = (const float*)d_in[0];
    const float* targ = (const float*)d_in[1];
    float* out = (float*)d_out;
    float* ws  = (float*)d_ws;  // NBLOCKS floats of scratch

    yolo_cell_kernel<<<NBLOCKS, 256, 0, stream>>>(pred, targ, ws, NCELLS);
    yolo_reduce_kernel<<<1, 256, 0, stream>>>(ws, NBLOCKS, out);
}